// ResidualVectorQuantizer_3513283248283
// MI455X (gfx1250) — compile-verified
//
#include <hip/hip_runtime.h>

// ---------------------------------------------------------------------------
// VQ codebook search on gfx1250 (MI455X):
//   distances = ||x||^2 + ||c||^2 - 2 x.cT ; codes = argmin ; q = codebook[codes]
// Compute-bound f32 GEMM -> V_WMMA_F32_16X16X4_F32 (f32 fidelity: argmin gaps
// ~1e-4 are too fragile for bf16 inputs). Codebook (8MB, L2-resident) streams
// through LDS via GLOBAL_LOAD_ASYNC_TO_LDS (ASYNCcnt) when available.
// Block = 128 thr = 4 waves: wave w -> rows (w&1)*16.., chunk parity w>>1.
// ---------------------------------------------------------------------------

#define N_ROWS   16384
#define KDIM     512
#define NCODES   4096
#define ROWSTRIDE 516   // 512 + 4 dword pad -> conflict-free ds_load_b64
#define BUFROWS  16     // codes per chunk (WMMA N)

typedef __attribute__((ext_vector_type(2))) float v2f;
typedef __attribute__((ext_vector_type(4))) float v4f;
typedef __attribute__((ext_vector_type(8))) float v8f;
typedef __attribute__((ext_vector_type(4))) int   v4i;

// ----- async global->LDS staging (CDNA5), with safe fallback ----------------
#if __has_builtin(__builtin_amdgcn_global_load_async_to_lds_b128)
#define HAVE_ASYNC_LDS 1
// Diagnostic from probe: param0 is 'v4i __attribute__((address_space(1)))*'
// (printed as '__device__' in HIP mode). Param1 is the LDS-side pointer (AS3).
typedef __attribute__((address_space(1))) v4i gv4i;
typedef __attribute__((address_space(3))) v4i lv4i;

__device__ __forceinline__ void async_copy16(const float* g, float* l) {
  // generic shared ptr low 32 bits == LDS byte offset (aperture rule)
  __builtin_amdgcn_global_load_async_to_lds_b128(
      (gv4i*)(__UINTPTR_TYPE__)(const void*)g,
      (lv4i*)(unsigned)(__UINTPTR_TYPE__)(void*)l,
      /*offset=*/0, /*cpol=*/0);
}
__device__ __forceinline__ void async_wait0() {
#if __has_builtin(__builtin_amdgcn_s_wait_asynccnt)
  __builtin_amdgcn_s_wait_asynccnt(0);
#else
  asm volatile("s_wait_asynccnt 0x0" ::: "memory");
#endif
}
#else
#define HAVE_ASYNC_LDS 0
#endif

__device__ __forceinline__ void stage16(const float* g, float* l) {
#if HAVE_ASYNC_LDS
  async_copy16(g, l);
#else
  *(v4f*)l = *(const v4f*)g;
#endif
}
__device__ __forceinline__ void stage_fence() {
#if HAVE_ASYNC_LDS
  async_wait0();
#endif
  __syncthreads();
}

// ---------------- kernel 0: codebook squared norms + zero the loss ----------
__global__ __launch_bounds__(256) void vq_cbnorm_kernel(
    const float* __restrict__ cb, float* __restrict__ cbnorm,
    float* __restrict__ loss) {
  const int row = blockIdx.x;
  const int t   = threadIdx.x;
  const float* r = cb + (size_t)row * KDIM;
  float s = 0.f;
  #pragma unroll
  for (int i = t; i < KDIM; i += 256) { float v = r[i]; s += v * v; }
  #pragma unroll
  for (int m = 16; m >= 1; m >>= 1) s += __shfl_xor(s, m, 32);
  __shared__ float ws[8];
  if ((t & 31) == 0) ws[t >> 5] = s;
  __syncthreads();
  if (t == 0) {
    float tot = 0.f;
    #pragma unroll
    for (int i = 0; i < 8; ++i) tot += ws[i];
    cbnorm[row] = tot;
    if (row == 0) *loss = 0.f;
  }
}

// ---------------- kernel 1: WMMA distance + running argmin ------------------
__global__ __launch_bounds__(128) void vq_argmin_kernel(
    const float* __restrict__ x, const float* __restrict__ cb,
    const float* __restrict__ cbnorm, int* __restrict__ codes_i,
    float* __restrict__ codes_f) {
  extern __shared__ float smem[];
  float* xs = smem;                      // 32 * ROWSTRIDE  (x tile)
  float* bs = smem + 32 * ROWSTRIDE;     // 2 * 16 * ROWSTRIDE (even|odd chunk)
  __shared__ float pbest[2][32];         // partial argmin merge across parities
  __shared__ int   pcode[2][32];

  const int tid    = threadIdx.x;
  const int lane   = tid & 31;
  const int wave   = tid >> 5;           // 0..3
  const int rowgrp = wave & 1;           // rows rowgrp*16 ..
  const int parity = wave >> 1;          // even(0)/odd(1) 16-code chunks
  const int half   = lane >> 4;
  const int lr     = lane & 15;
  const int row0   = blockIdx.x * 32;

  // Stage x tile (32 rows x 512 f32) into padded LDS.
  for (int i = tid; i < 32 * 128; i += 128) {
    const int r = i >> 7, c4 = i & 127;
    stage16(x + (size_t)(row0 + r) * KDIM + c4 * 4, xs + r * ROWSTRIDE + c4 * 4);
  }

  // A frag (16x4): lane<16 -> M=lane,K={k,k+1}; lane>=16 -> K={k+2,k+3}
  // B frag (4x16): lane<16 -> N=lane,K={k,k+1}; lane>=16 -> K={k+2,k+3}
  const float* xrow_lds = xs + (rowgrp * 16 + lr) * ROWSTRIDE + half * 2;
  const float* brow_lds = bs + (parity * BUFROWS + lr) * ROWSTRIDE + half * 2;

  float best[8];
  int   bcode[8];
  #pragma unroll
  for (int v = 0; v < 8; ++v) { best[v] = 3.4e38f; bcode[v] = 0; }

  for (int t = 0; t < NCODES / 32; ++t) {
    __syncthreads();                     // previous chunk fully consumed
    // Stage 32 codes: rows 0..15 -> even buffer, 16..31 -> odd (contiguous).
    for (int i = tid; i < 32 * 128; i += 128) {
      const int r = i >> 7, c4 = i & 127;
      stage16(cb + (size_t)(32 * t + r) * KDIM + c4 * 4,
              bs + r * ROWSTRIDE + c4 * 4);
    }
    stage_fence();                       // s_wait_asynccnt 0 + barrier

    const int code = 32 * t + parity * 16 + lr;
    const float cn = cbnorm[code];       // ||c||^2 for this lane's column

    // Two interleaved accumulator chains -> back-to-back WMMA issue.
    v8f acc0 = {0.f,0.f,0.f,0.f,0.f,0.f,0.f,0.f};
    v8f acc1 = {0.f,0.f,0.f,0.f,0.f,0.f,0.f,0.f};
    #pragma unroll 8
    for (int k = 0; k < KDIM; k += 8) {
      v2f a0 = *(const v2f*)(xrow_lds + k);
      v2f b0 = *(const v2f*)(brow_lds + k);
      v2f a1 = *(const v2f*)(xrow_lds + k + 4);
      v2f b1 = *(const v2f*)(brow_lds + k + 4);
      acc0 = __builtin_amdgcn_wmma_f32_16x16x4_f32(false, a0, false, b0,
                                                   (short)0, acc0, false, false);
      acc1 = __builtin_amdgcn_wmma_f32_16x16x4_f32(false, a1, false, b1,
                                                   (short)0, acc1, false, false);
    }

    // score = ||c||^2 - 2 x.c  (||x||^2 row-constant -> argmin-invariant)
    #pragma unroll
    for (int v = 0; v < 8; ++v) {
      float s = fmaf(-2.f, acc0[v] + acc1[v], cn);
      if (s < best[v]) { best[v] = s; bcode[v] = code; }  // strict <: first wins
    }
  }

  // Min-reduce across the 16 lanes of each half, then publish per-parity.
  #pragma unroll
  for (int v = 0; v < 8; ++v) {
    #pragma unroll
    for (int m = 8; m >= 1; m >>= 1) {
      float os = __shfl_xor(best[v], m, 32);
      int   oc = __shfl_xor(bcode[v], m, 32);
      if (os < best[v] || (os == best[v] && oc < bcode[v])) {
        best[v] = os; bcode[v] = oc;
      }
    }
    if (lr == v) {                       // C layout: M = v + 8*half
      const int rl = rowgrp * 16 + v + half * 8;
      pbest[parity][rl] = best[v];
      pcode[parity][rl] = bcode[v];
    }
  }
  __syncthreads();

  // Merge even/odd-parity candidates; tie-break to lowest code (jnp.argmin).
  if (tid < 32) {
    float b0 = pbest[0][tid], b1 = pbest[1][tid];
    int   c0 = pcode[0][tid], c1 = pcode[1][tid];
    const bool take1 = (b1 < b0) || (b1 == b0 && c1 < c0);
    const int row = row0 + tid;
    const int c   = take1 ? c1 : c0;
    codes_i[row] = c;
    codes_f[row] = (float)c;
  }
}

// ---------------- kernel 2: gather quantized rows + loss --------------------
__global__ __launch_bounds__(256) void vq_gather_loss_kernel(
    const float* __restrict__ x, const float* __restrict__ cb,
    const int* __restrict__ codes_i, float* __restrict__ outq,
    float* __restrict__ loss) {
  const int row = blockIdx.x;
  const int t   = threadIdx.x;
  const int code = codes_i[row];
  const float* q  = cb + (size_t)code * KDIM;
  const float* xr = x  + (size_t)row  * KDIM;
  float s = 0.f;
  #pragma unroll
  for (int i = t * 2; i < KDIM; i += 512) {
    v2f qv = *(const v2f*)(q + i);
    v2f xv = *(const v2f*)(xr + i);
    *(v2f*)(outq + (size_t)row * KDIM + i) = qv;   // x + sg(q-x) == q
    const float d0 = qv.x - xv.x, d1 = qv.y - xv.y;
    s += d0 * d0 + d1 * d1;
  }
  #pragma unroll
  for (int m = 16; m >= 1; m >>= 1) s += __shfl_xor(s, m, 32);
  __shared__ float ws[8];
  if ((t & 31) == 0) ws[t >> 5] = s;
  __syncthreads();
  if (t == 0) {
    float tot = 0.f;
    #pragma unroll
    for (int i = 0; i < 8; ++i) tot += ws[i];
    // loss = mean((sg(q)-x)^2) + mean((q-sg(x))^2) = 2*mean((q-x)^2)
    atomicAdd(loss, tot * (2.0f / ((float)N_ROWS * (float)KDIM)));
  }
}

// ---------------------------------------------------------------------------
extern "C" void kernel_launch(void* const* d_in, const int* in_sizes, int n_in,
                              void* d_out, int out_size, void* d_ws, size_t ws_size,
                              hipStream_t stream) {
  (void)in_sizes; (void)n_in; (void)out_size; (void)ws_size;
  const float* x  = (const float*)d_in[0];
  const float* cb = (const float*)d_in[1];

  float* out_q     = (float*)d_out;                       // [N, 512]
  float* out_codes = out_q + (size_t)N_ROWS * KDIM;       // [N] (as float)
  float* out_loss  = out_codes + N_ROWS;                  // [1]

  float* cbnorm  = (float*)d_ws;                          // 4096 f32
  int*   codes_i = (int*)((float*)d_ws + NCODES);         // 16384 i32

  vq_cbnorm_kernel<<<NCODES, 256, 0, stream>>>(cb, cbnorm, out_loss);

  const size_t lds_bytes = (size_t)(32 + 32) * ROWSTRIDE * sizeof(float); // ~129KB
  vq_argmin_kernel<<<N_ROWS / 32, 128, lds_bytes, stream>>>(
      x, cb, cbnorm, codes_i, out_codes);

  vq_gather_loss_kernel<<<N_ROWS, 256, 0, stream>>>(x, cb, codes_i, out_q, out_loss);
}